// GDNWithForecasting_28784870818517
// MI455X (gfx1250) — compile-verified
//
#include <hip/hip_runtime.h>

#define B_   4096
#define N_   128
#define H_   128
#define W_   64
#define O_   384   // NHOR * N
#define XS   132   // padded LDS row stride in floats (conflict-free: 132 % 64 = 4)

typedef __attribute__((ext_vector_type(2))) float v2f;
typedef __attribute__((ext_vector_type(8))) float v8f;

__device__ __forceinline__ v8f wmma_f32(v2f a, v2f b, v8f c) {
  // 8 args: (neg_a, A, neg_b, B, c_mod, C, reuse_a, reuse_b)
  return __builtin_amdgcn_wmma_f32_16x16x4_f32(false, a, false, b, (short)0, c,
                                               false, false);
}

// A-fragment (16x4 f32): lanes 0-15 hold (M=l, K=k..k+1), lanes 16-31 (M=l, K=k+2..k+3)
__device__ __forceinline__ v2f load_a(const float* Xs, int m0, int k, int l15, int lhalf) {
  const float* p = Xs + (m0 + l15) * XS + k + lhalf * 2;
  v2f a; a.x = p[0]; a.y = p[1];
  return a;
}

// B-fragment (4x16 f32) from row-major W[k][col], leading dim ldw
__device__ __forceinline__ v2f load_b(const float* W, int k, int n0, int ldw,
                                      int l15, int lhalf) {
  const float* q = W + (size_t)(k + lhalf * 2) * ldw + n0 + l15;
  v2f b; b.x = q[0]; b.y = q[ldw];
  return b;
}

// B-fragment from a transposed weight: B[k][col] = Wt[col*ldw + k]
__device__ __forceinline__ v2f load_bt(const float* Wt, int k, int n0, int ldw,
                                       int l15, int lhalf) {
  const float* q = Wt + (size_t)(n0 + l15) * ldw + k + lhalf * 2;
  v2f b; b.x = q[0]; b.y = q[1];
  return b;
}

// ---------------------------------------------------------------------------
// Forecast head: h = relu(emb @ fc1_w^T + fc1_b); out = h @ fc2_w^T + fc2_b
// One block per 32-row B tile. 8 waves.
// ---------------------------------------------------------------------------
__global__ __launch_bounds__(256) void forecast_kernel(
    const float* __restrict__ emb, const float* __restrict__ fc1w,
    const float* __restrict__ fc1b, const float* __restrict__ fc2w,
    const float* __restrict__ fc2b, float* __restrict__ out) {
  __shared__ float Xs[32 * XS];
  __shared__ float Hs[32 * XS];
  const int tid = threadIdx.x;
  const int wave = tid >> 5, lane = tid & 31;
  const int l15 = lane & 15, lhalf = lane >> 4;
  const int b0 = blockIdx.x * 32;

  // Stage embeddings tile (32 x 128) into LDS.
#pragma unroll
  for (int i = 0; i < 4; ++i) {
    int idx = tid + i * 256;
    int row = idx >> 5, c4 = (idx & 31) * 4;
    float4 v = *(const float4*)(emb + (size_t)(b0 + row) * H_ + c4);
    *(float4*)(&Xs[row * XS + c4]) = v;
  }
  __syncthreads();

  // GEMM1: 32x128, each wave one 16x32 strip (2 accumulators).
  const int m0 = (wave >> 2) * 16;
  const int n0 = (wave & 3) * 32;
  v8f acc0 = {}; v8f acc1 = {};
  for (int k = 0; k < H_; k += 4) {
    v2f a  = load_a(Xs, m0, k, l15, lhalf);
    v2f bA = load_bt(fc1w, k, n0,      H_, l15, lhalf);
    v2f bB = load_bt(fc1w, k, n0 + 16, H_, l15, lhalf);
    acc0 = wmma_f32(a, bA, acc0);
    acc1 = wmma_f32(a, bB, acc1);
  }
#pragma unroll
  for (int r = 0; r < 8; ++r) {
    int row = m0 + r + lhalf * 8;
    int c = n0 + l15;
    Hs[row * XS + c]      = fmaxf(acc0[r] + fc1b[c], 0.0f);
    Hs[row * XS + c + 16] = fmaxf(acc1[r] + fc1b[c + 16], 0.0f);
  }
  __syncthreads();

  // GEMM2: 32x384, each wave a 16x96 strip (6 accumulators).
  const int o0 = (wave & 3) * 96;
  v8f acc[6] = {};
  for (int k = 0; k < H_; k += 4) {
    v2f a = load_a(Hs, m0, k, l15, lhalf);
#pragma unroll
    for (int t = 0; t < 6; ++t) {
      v2f b = load_bt(fc2w, k, o0 + t * 16, H_, l15, lhalf);
      acc[t] = wmma_f32(a, b, acc[t]);
    }
  }
#pragma unroll
  for (int t = 0; t < 6; ++t) {
#pragma unroll
    for (int r = 0; r < 8; ++r) {
      int row = m0 + r + lhalf * 8;
      int o = o0 + t * 16 + l15;
      out[(size_t)(b0 + row) * O_ + o] = acc[t][r] + fc2b[o];
    }
  }
}

// ---------------------------------------------------------------------------
// Reconstruction heads: per-node MLP, grid = (node, B/64). 64-row B tiles,
// single LDS buffer reused for X then H (GEMM1 fully consumes X before the
// barrier, so H may overwrite it). 8 waves = 8 column tiles: rw1[n] is read
// exactly once per block.
// hn = relu(X @ rw1[n] + rb1[n]); out[b, w, n] = (hn @ rw2[n] + rb2[n])[b, w]
// ---------------------------------------------------------------------------
__global__ __launch_bounds__(256) void recon_kernel(
    const float* __restrict__ se, const float* __restrict__ rw1,
    const float* __restrict__ rb1, const float* __restrict__ rw2,
    const float* __restrict__ rb2, float* __restrict__ out) {
  __shared__ float S[64 * XS];
  const int tid = threadIdx.x;
  const int wave = tid >> 5, lane = tid & 31;
  const int l15 = lane & 15, lhalf = lane >> 4;
  const int node = blockIdx.x;
  const int b0 = blockIdx.y * 64;

  // Stage sensor_embeddings[b0..b0+63, node, :] (64 x 128) into LDS.
  const float* xg = se + (size_t)node * H_;
#pragma unroll
  for (int i = 0; i < 8; ++i) {
    int idx = tid + i * 256;
    int row = idx >> 5, c4 = (idx & 31) * 4;
    float4 v = *(const float4*)(xg + (size_t)(b0 + row) * N_ * H_ + c4);
    *(float4*)(&S[row * XS + c4]) = v;
  }
  __syncthreads();

  // GEMM1: 64x128 = relu(X @ rw1[n] + rb1[n]).
  // Wave w owns column tile n0 = w*16 and all four 16-row M tiles, so each
  // B-fragment is loaded once per block and amortized over 4 WMMAs.
  const float* W1 = rw1 + (size_t)node * H_ * H_;
  const float* bias1 = rb1 + (size_t)node * H_;
  const int n0 = (wave & 7) * 16;
  v8f acc[4] = {};
  for (int k = 0; k < H_; k += 4) {
    v2f b = load_b(W1, k, n0, H_, l15, lhalf);
#pragma unroll
    for (int t = 0; t < 4; ++t) {
      v2f a = load_a(S, t * 16, k, l15, lhalf);
      acc[t] = wmma_f32(a, b, acc[t]);
    }
  }
  __syncthreads();  // all waves finished reading X
#pragma unroll
  for (int t = 0; t < 4; ++t) {
#pragma unroll
    for (int r = 0; r < 8; ++r) {
      int row = t * 16 + r + lhalf * 8;
      int c = n0 + l15;
      S[row * XS + c] = fmaxf(acc[t][r] + bias1[c], 0.0f);
    }
  }
  __syncthreads();

  // GEMM2: 64x64 = H @ rw2[n]. Wave w: column tile (w&3), M tiles {2(w>>2), 2(w>>2)+1}.
  const float* W2 = rw2 + (size_t)node * H_ * W_;
  const float* bias2 = rb2 + (size_t)node * W_;
  const int n2 = (wave & 3) * 16;
  const int mb = (wave >> 2) * 32;
  v8f acc2[2] = {};
  for (int k = 0; k < H_; k += 4) {
    v2f b = load_b(W2, k, n2, W_, l15, lhalf);
#pragma unroll
    for (int t = 0; t < 2; ++t) {
      v2f a = load_a(S, mb + t * 16, k, l15, lhalf);
      acc2[t] = wmma_f32(a, b, acc2[t]);
    }
  }
  // Fused bias + transpose store: out[b, w, n].
#pragma unroll
  for (int t = 0; t < 2; ++t) {
#pragma unroll
    for (int r = 0; r < 8; ++r) {
      int row = mb + t * 16 + r + lhalf * 8;
      int c = n2 + l15;
      out[(size_t)(b0 + row) * (W_ * N_) + (size_t)c * N_ + node] = acc2[t][r] + bias2[c];
    }
  }
}

// ---------------------------------------------------------------------------
// Logits passthrough.
// ---------------------------------------------------------------------------
__global__ __launch_bounds__(256) void copy_logits(const float* __restrict__ in,
                                                   float* __restrict__ out, int n4) {
  int i = blockIdx.x * blockDim.x + threadIdx.x;
  if (i < n4) ((float4*)out)[i] = ((const float4*)in)[i];
}

extern "C" void kernel_launch(void* const* d_in, const int* in_sizes, int n_in,
                              void* d_out, int out_size, void* d_ws, size_t ws_size,
                              hipStream_t stream) {
  const float* emb  = (const float*)d_in[0];
  const float* se   = (const float*)d_in[1];
  const float* slog = (const float*)d_in[2];
  const float* fc1w = (const float*)d_in[3];
  const float* fc1b = (const float*)d_in[4];
  const float* fc2w = (const float*)d_in[5];
  const float* fc2b = (const float*)d_in[6];
  const float* rw1  = (const float*)d_in[7];
  const float* rb1  = (const float*)d_in[8];
  const float* rw2  = (const float*)d_in[9];
  const float* rb2  = (const float*)d_in[10];

  float* out       = (float*)d_out;
  float* out_fore  = out + (size_t)B_ * N_;          // forecast (B, NHOR*N)
  float* out_recon = out_fore + (size_t)B_ * O_;     // reconstruction (B, W, N)

  const int n4 = (B_ * N_) / 4;
  copy_logits<<<(n4 + 255) / 256, 256, 0, stream>>>(slog, out, n4);
  forecast_kernel<<<B_ / 32, 256, 0, stream>>>(emb, fc1w, fc1b, fc2w, fc2b, out_fore);
  recon_kernel<<<dim3(N_, B_ / 64), 256, 0, stream>>>(se, rw1, rb1, rw2, rb2, out_recon);
}